// Net_2018634629806
// MI455X (gfx1250) — compile-verified
//
#include <hip/hip_runtime.h>
#include <hip/hip_bf16.h>

// ---------------- problem constants (match reference) ----------------
#define NN 100000          // nodes
#define NE 1600000         // edges
#define CIN 768
#define CHID 128
#define COUT 64

typedef float v2f __attribute__((ext_vector_type(2)));
typedef float v8f __attribute__((ext_vector_type(8)));

// ---------------- degree / norm ----------------
__global__ __launch_bounds__(256) void deg_init_kernel(float* __restrict__ deg) {
    int i = blockIdx.x * 256 + threadIdx.x;
    if (i < NN) deg[i] = 1.0f;                       // self-loop
}

__global__ __launch_bounds__(256) void deg_scatter_kernel(const int* __restrict__ dst,
                                                          float* __restrict__ deg) {
    int e = blockIdx.x * 256 + threadIdx.x;
    if (e < NE) unsafeAtomicAdd(&deg[dst[e]], 1.0f);
}

__global__ __launch_bounds__(256) void dis_kernel(float* __restrict__ deg) {
    int i = blockIdx.x * 256 + threadIdx.x;
    if (i < NN) deg[i] = __frsqrt_rn(deg[i]);        // deg >= 1 always
}

// ---------------- GEMM1: xw = x @ W1  (fp32 WMMA 16x16x4) ----------------
// block = 256 thr (8 waves). Block owns 16 rows; wave w owns cols [16w, 16w+16).
#define LDSTRIDE1 772   // 768 + 4 pad -> conflict-free A-fragment reads
__global__ __launch_bounds__(256)
void gemm1_kernel(const float* __restrict__ x, const float* __restrict__ W1,
                  float* __restrict__ xw) {
    __shared__ float tileA[16 * LDSTRIDE1];
    const int row0 = blockIdx.x * 16;

    // cooperative, coalesced load of 16x768 x-tile
    for (int r = 0; r < 16; ++r)
        for (int c = threadIdx.x; c < CIN; c += 256)
            tileA[r * LDSTRIDE1 + c] = x[(size_t)(row0 + r) * CIN + c];
    __syncthreads();

    const int wave = threadIdx.x >> 5;
    const int lane = threadIdx.x & 31;
    const int col0 = wave * 16;
    const int m    = lane & 15;     // M index (A) == N index (B/D) for this lane
    const int half = lane >> 4;     // 0: K=0,1   1: K=2,3
    const int kOff = half * 2;

    v8f acc = {0.f, 0.f, 0.f, 0.f, 0.f, 0.f, 0.f, 0.f};
    const float* ap = &tileA[m * LDSTRIDE1 + kOff];
    const float* bp = &W1[(size_t)kOff * CHID + col0 + m];

    for (int kb = 0; kb < CIN; kb += 4) {
        v2f a; a.x = ap[kb];            a.y = ap[kb + 1];
        v2f b; b.x = bp[kb * CHID];     b.y = bp[kb * CHID + CHID];
        acc = __builtin_amdgcn_wmma_f32_16x16x4_f32(
                  false, a, false, b, (short)0, acc, false, false);
    }

    // D layout: VGPR r -> M = r + 8*half, N = m
    float* op = &xw[(size_t)(row0 + half * 8) * CHID + col0 + m];
#pragma unroll
    for (int r = 0; r < 8; ++r) op[r * CHID] = acc[r];
}

// ---------------- layer-1 aggregation ----------------
__global__ __launch_bounds__(256)
void init_h_kernel(const float* __restrict__ xw, const float* __restrict__ dis,
                   const float* __restrict__ b1, float* __restrict__ h) {
    int i = blockIdx.x * 256 + threadIdx.x;        // i < NN*CHID (exact grid)
    int n = i >> 7, c = i & 127;
    float di = dis[n];
    h[i] = di * di * xw[i] + b1[c];                // self-loop + bias folded in
}

__global__ __launch_bounds__(256)
void scatter1_kernel(const int* __restrict__ src, const int* __restrict__ dst,
                     const float* __restrict__ dis, const float* __restrict__ xw,
                     float* __restrict__ h) {
    const int wave = threadIdx.x >> 5, lane = threadIdx.x & 31;
    const int e = blockIdx.x * 8 + wave;           // NE/8 blocks exactly
    const int s = src[e], d = dst[e];
    const float w = dis[s] * dis[d];
    const float4 v = *(const float4*)&xw[(size_t)s * CHID + lane * 4];
    float* hp = &h[(size_t)d * CHID + lane * 4];
    unsafeAtomicAdd(hp + 0, w * v.x);
    unsafeAtomicAdd(hp + 1, w * v.y);
    unsafeAtomicAdd(hp + 2, w * v.z);
    unsafeAtomicAdd(hp + 3, w * v.w);
}

// ---------------- GEMM2: xw2 = relu(h) @ W2  (fp32 WMMA) ----------------
// block = 256 thr (8 waves). Block owns 32 rows; wave -> (rowtile, coltile).
#define LDSTRIDE2 132   // 128 + 4 pad
__global__ __launch_bounds__(256)
void gemm2_kernel(const float* __restrict__ h, const float* __restrict__ W2,
                  float* __restrict__ xw2) {
    __shared__ float tileA[32 * LDSTRIDE2];
    const int row0 = blockIdx.x * 32;

    for (int i = threadIdx.x; i < 32 * CHID; i += 256) {
        int r = i >> 7, c = i & 127;
        tileA[r * LDSTRIDE2 + c] = fmaxf(h[(size_t)(row0 + r) * CHID + c], 0.f); // ReLU fused
    }
    __syncthreads();

    const int wave = threadIdx.x >> 5;
    const int lane = threadIdx.x & 31;
    const int rt   = wave >> 2;          // row tile 0..1
    const int ct   = wave & 3;           // col tile 0..3
    const int col0 = ct * 16;
    const int m    = lane & 15;
    const int half = lane >> 4;
    const int kOff = half * 2;

    v8f acc = {0.f, 0.f, 0.f, 0.f, 0.f, 0.f, 0.f, 0.f};
    const float* ap = &tileA[(rt * 16 + m) * LDSTRIDE2 + kOff];
    const float* bp = &W2[(size_t)kOff * COUT + col0 + m];

    for (int kb = 0; kb < CHID; kb += 4) {
        v2f a; a.x = ap[kb];            a.y = ap[kb + 1];
        v2f b; b.x = bp[kb * COUT];     b.y = bp[kb * COUT + COUT];
        acc = __builtin_amdgcn_wmma_f32_16x16x4_f32(
                  false, a, false, b, (short)0, acc, false, false);
    }

    float* op = &xw2[(size_t)(row0 + rt * 16 + half * 8) * COUT + col0 + m];
#pragma unroll
    for (int r = 0; r < 8; ++r) op[r * COUT] = acc[r];
}

// ---------------- layer-2 aggregation (into d_out) ----------------
__global__ __launch_bounds__(256)
void init_z_kernel(const float* __restrict__ xw2, const float* __restrict__ dis,
                   const float* __restrict__ b2, float* __restrict__ z) {
    int i = blockIdx.x * 256 + threadIdx.x;        // i < NN*COUT (exact grid)
    int n = i >> 6, c = i & 63;
    float di = dis[n];
    z[i] = di * di * xw2[i] + b2[c];
}

__global__ __launch_bounds__(256)
void scatter2_kernel(const int* __restrict__ src, const int* __restrict__ dst,
                     const float* __restrict__ dis, const float* __restrict__ xw2,
                     float* __restrict__ z) {
    const int wave = threadIdx.x >> 5, lane = threadIdx.x & 31;
    const int e = blockIdx.x * 8 + wave;
    const int s = src[e], d = dst[e];
    const float w = dis[s] * dis[d];
    const float2 v = *(const float2*)&xw2[(size_t)s * COUT + lane * 2];
    float* zp = &z[(size_t)d * COUT + lane * 2];
    unsafeAtomicAdd(zp + 0, w * v.x);
    unsafeAtomicAdd(zp + 1, w * v.y);
}

// ---------------- launch ----------------
extern "C" void kernel_launch(void* const* d_in, const int* in_sizes, int n_in,
                              void* d_out, int out_size, void* d_ws, size_t ws_size,
                              hipStream_t stream) {
    const float* x  = (const float*)d_in[0];
    const int*   ei = (const int*)d_in[1];       // [2, E] flat
    const float* W1 = (const float*)d_in[2];
    const float* b1 = (const float*)d_in[3];
    const float* W2 = (const float*)d_in[4];
    const float* b2 = (const float*)d_in[5];
    const int* src = ei;
    const int* dst = ei + NE;
    float* out = (float*)d_out;

    // workspace layout: [dis: 512KB pad][xw: N*128 f32][h: N*128 f32]; xw2 aliases xw
    char* ws = (char*)d_ws;
    float* dis = (float*)ws;
    float* xw  = (float*)(ws + (1 << 19));
    float* h   = xw + (size_t)NN * CHID;
    float* xw2 = xw;                                 // safe: xw dead after scatter1

    deg_init_kernel   <<<(NN + 255) / 256, 256, 0, stream>>>(dis);
    deg_scatter_kernel<<<(NE + 255) / 256, 256, 0, stream>>>(dst, dis);
    dis_kernel        <<<(NN + 255) / 256, 256, 0, stream>>>(dis);

    gemm1_kernel      <<<NN / 16, 256, 0, stream>>>(x, W1, xw);
    init_h_kernel     <<<(NN * CHID) / 256, 256, 0, stream>>>(xw, dis, b1, h);
    scatter1_kernel   <<<NE / 8, 256, 0, stream>>>(src, dst, dis, xw, h);

    gemm2_kernel      <<<NN / 32, 256, 0, stream>>>(h, W2, xw2);
    init_z_kernel     <<<(NN * COUT) / 256, 256, 0, stream>>>(xw2, dis, b2, out);
    scatter2_kernel   <<<NE / 8, 256, 0, stream>>>(src, dst, dis, xw2, out);
}